// C_LNCCLoss_80788334838284
// MI455X (gfx1250) — compile-verified
//
#include <hip/hip_runtime.h>
#include <math.h>

typedef float v2f __attribute__((ext_vector_type(2)));
typedef float v8f __attribute__((ext_vector_type(8)));

#define IMG_H   512
#define IMG_W   512
#define NBATCH  16
#define TILE    16
#define HALO    4
#define HROWS   24            // TILE + 2*HALO
#define HSTRIDE 17            // padded row stride (16 cols + 1) -> conflict-free banks
#define HFIELD  (HROWS * HSTRIDE)   // 408 floats per field
#define NFIELDS 10
#define WAVE_LDS (NFIELDS * HFIELD) // 4080 floats per wave
#define WAVES   4
#define NBLOCKS 1024
#define ITERS   4             // tiles per wave; 1024*4*4 = 16384 tiles total

__device__ __forceinline__ float lncc_val(float s1, float s2, float q1, float q2, float s12) {
    // faithful to reference: var = clip(localsum(x^2) - mean^2, 0), cov = localsum(xy) - mean1*mean2
    const float inv_n = 1.0f / 81.0f;
    float mu1 = s1 * inv_n;
    float mu2 = s2 * inv_n;
    float v1 = q1 - mu1 * mu1; v1 = v1 > 0.0f ? v1 : 0.0f;
    float v2 = q2 - mu2 * mu2; v2 = v2 > 0.0f ? v2 : 0.0f;
    float cov = s12 - mu1 * mu2;
    return cov / sqrtf((v1 + 1e-5f) * (v2 + 1e-5f));
}

__global__ __launch_bounds__(WAVES * 32)
void lncc_tiles(const float* __restrict__ pred,
                const float* __restrict__ target,
                const float* __restrict__ mask,
                float2* __restrict__ partials) {
    __shared__ float HB[WAVES][WAVE_LDS];

    const int lane = threadIdx.x & 31;
    const int w    = threadIdx.x >> 5;
    const int half = lane >> 4;     // 0 or 1 (wave32 lane halves)
    const int nn   = lane & 15;     // M for the A matrix, N for B/C/D

    // Constant band matrix A (16x24), K split into 6 chunks of 4.
    // OUT[M][N] = sum_{r=M}^{M+8} H[r][N]  =>  Band[M][r] = (M <= r <= M+8)
    // A 16x4 layout: lane holds row M=lane&15; VGPR0/1 = K 0,1 (lanes 0-15) or 2,3 (lanes 16-31)
    v2f A[6];
#pragma unroll
    for (int k = 0; k < 6; ++k) {
        int kb = 4 * k + 2 * half;
        A[k].x = (kb     >= nn && kb     <= nn + 8) ? 1.0f : 0.0f;
        A[k].y = (kb + 1 >= nn && kb + 1 <= nn + 8) ? 1.0f : 0.0f;
    }

    float sp_acc = 0.0f, sn_acc = 0.0f;

    for (int iter = 0; iter < ITERS; ++iter) {
        int tile = iter * (NBLOCKS * WAVES) + blockIdx.x * WAVES + w;
        int b    = tile >> 10;          // 1024 tiles per image
        int rem  = tile & 1023;
        int y0   = (rem >> 5) << 4;     // 32 tiles per row
        int x0   = (rem & 31) << 4;

        float* Hw = &HB[w][0];

        // ---------- horizontal pass: lanes 0..23 each own one input row ----------
        if (lane < HROWS) {
            int  r     = y0 - HALO + lane;
            bool rowok = (r >= 0) && (r < IMG_H);
            int  rc    = rowok ? r : 0;
            const float* prow  = pred   + ((size_t)b * IMG_H + rc) * IMG_W;
            const float* trow  = target + ((size_t)b * IMG_H + rc) * IMG_W;
            const float* m0row = mask   + (((size_t)b * 2 + 0) * IMG_H + rc) * IMG_W;
            const float* m1row = m0row + (size_t)IMG_H * IMG_W;

            float s0=0,s1=0,s2=0,s3=0,s4=0,s5=0,s6=0,s7=0,s8=0,s9=0;

            auto accum = [&](int x, float sgn) {
                if (!rowok || x < 0 || x >= IMG_W) return;
                float p  = prow[x];
                float t  = trow[x];
                float mm = (m1row[x] > m0row[x]) ? 1.0f : 0.0f;  // argmax over 2 channels
                float pp = p * p, tt = t * t, pt = p * t;
                s0 += sgn * p;  s1 += sgn * t;  s2 += sgn * pp;
                s3 += sgn * tt; s4 += sgn * pt;
                float sm = sgn * mm;
                s5 += sm * p;  s6 += sm * t;  s7 += sm * pp;
                s8 += sm * tt; s9 += sm * pt;
            };
            auto storeH = [&](int c) {
                float* hst = Hw + lane * HSTRIDE + c;
                hst[0*HFIELD]=s0; hst[1*HFIELD]=s1; hst[2*HFIELD]=s2; hst[3*HFIELD]=s3;
                hst[4*HFIELD]=s4; hst[5*HFIELD]=s5; hst[6*HFIELD]=s6; hst[7*HFIELD]=s7;
                hst[8*HFIELD]=s8; hst[9*HFIELD]=s9;
            };

#pragma unroll
            for (int dx = -HALO; dx <= HALO; ++dx) accum(x0 + dx, 1.0f);
            storeH(0);
            for (int c = 1; c < TILE; ++c) {      // sliding window: +entering tap, -leaving tap
                accum(x0 + c + HALO,     1.0f);
                accum(x0 + c - HALO - 1, -1.0f);
                storeH(c);
            }
        }
        __syncthreads();

        // ---------- vertical pass: 6-chain of V_WMMA_F32_16X16X4_F32 per field ----------
        v8f S[NFIELDS];
#pragma unroll
        for (int f = 0; f < NFIELDS; ++f) {
            const float* Hf = Hw + f * HFIELD;
            v8f c = {0.0f,0.0f,0.0f,0.0f,0.0f,0.0f,0.0f,0.0f};
#pragma unroll
            for (int k = 0; k < 6; ++k) {
                // B 4x16 layout: VGPR0 = row 4k (lanes 0-15) / 4k+2 (lanes 16-31); VGPR1 = +1
                int r0 = 4 * k + 2 * half;
                v2f bb;
                bb.x = Hf[ r0      * HSTRIDE + nn];
                bb.y = Hf[(r0 + 1) * HSTRIDE + nn];
                c = __builtin_amdgcn_wmma_f32_16x16x4_f32(
                        /*neg_a=*/false, A[k], /*neg_b=*/false, bb,
                        /*c_mod=*/(short)0, c, /*reuse_a=*/false, /*reuse_b=*/false);
            }
            S[f] = c;   // S[f][j] = boxsum at pixel (y0 + j + 8*half', x0 + nn)
        }
        __syncthreads();

        // ---------- pointwise LNCC; unmasked branch = full - masked ----------
#pragma unroll
        for (int j = 0; j < 8; ++j) {
            float t0=S[0][j], t1=S[1][j], t2=S[2][j], t3=S[3][j], t4=S[4][j];
            float n0=S[5][j], n1=S[6][j], n2=S[7][j], n3=S[8][j], n4=S[9][j];
            sn_acc += lncc_val(n0, n1, n2, n3, n4);
            sp_acc += lncc_val(t0-n0, t1-n1, t2-n2, t3-n3, t4-n4);
        }
    }

    // ---------- deterministic reduction: wave butterfly, then per-block sum ----------
#pragma unroll
    for (int off = 16; off > 0; off >>= 1) {
        sp_acc += __shfl_xor(sp_acc, off, 32);
        sn_acc += __shfl_xor(sn_acc, off, 32);
    }
    __syncthreads();
    if (lane == 0) { HB[w][0] = sp_acc; HB[w][1] = sn_acc; }
    __syncthreads();
    if (threadIdx.x == 0) {
        float a = 0.0f, c = 0.0f;
        for (int i = 0; i < WAVES; ++i) { a += HB[i][0]; c += HB[i][1]; }
        partials[blockIdx.x] = make_float2(a, c);
    }
}

__global__ void lncc_finalize(const float2* __restrict__ partials, int n,
                              float* __restrict__ out) {
    __shared__ float rp[256];
    __shared__ float rn[256];
    int t = threadIdx.x;
    float sp = 0.0f, sn = 0.0f;
    for (int i = t; i < n; i += 256) {  // fixed-order strided partial sums
        float2 v = partials[i];
        sp += v.x; sn += v.y;
    }
    rp[t] = sp; rn[t] = sn;
    __syncthreads();
    for (int off = 128; off > 0; off >>= 1) {
        if (t < off) { rp[t] += rp[t + off]; rn[t] += rn[t + off]; }
        __syncthreads();
    }
    if (t == 0) {
        const float invN = 1.0f / (float)((size_t)NBATCH * IMG_H * IMG_W);
        // 0.2*(1-mean_p) - 0.8*(1-mean_n) = -0.6 - 0.2*mean_p + 0.8*mean_n
        out[0] = -0.6f - 0.2f * (rp[0] * invN) + 0.8f * (rn[0] * invN);
    }
}

extern "C" void kernel_launch(void* const* d_in, const int* in_sizes, int n_in,
                              void* d_out, int out_size, void* d_ws, size_t ws_size,
                              hipStream_t stream) {
    const float* pred   = (const float*)d_in[0];
    const float* target = (const float*)d_in[1];
    const float* mask   = (const float*)d_in[2];
    float2* partials    = (float2*)d_ws;   // NBLOCKS float2 = 8 KB

    lncc_tiles<<<NBLOCKS, WAVES * 32, 0, stream>>>(pred, target, mask, partials);
    lncc_finalize<<<1, 256, 0, stream>>>(partials, NBLOCKS, (float*)d_out);
}